// TransLayer_66159676227648
// MI455X (gfx1250) — compile-verified
//
#include <hip/hip_runtime.h>
#include <math.h>
#include <stdint.h>

typedef float v2f __attribute__((ext_vector_type(2)));
typedef float v8f __attribute__((ext_vector_type(8)));

#define HEADS 8
#define DH 16
#define DMODEL 128
#define NTOK 8192
#define BATCH 8
#define LM 64
#define GRP 128
#define KSZ 33
#define BH (BATCH*HEADS)
#define ROWS (BATCH*NTOK)

__device__ __forceinline__ v8f wmma4(v2f a, v2f b, v8f c) {
  return __builtin_amdgcn_wmma_f32_16x16x4_f32(false, a, false, b, (short)0, c, false, false);
}

__device__ __forceinline__ float rmax16(float v) {
  v = fmaxf(v, __shfl_xor(v, 1, 32));
  v = fmaxf(v, __shfl_xor(v, 2, 32));
  v = fmaxf(v, __shfl_xor(v, 4, 32));
  v = fmaxf(v, __shfl_xor(v, 8, 32));
  return v;
}
__device__ __forceinline__ float rsum16(float v) {
  v += __shfl_xor(v, 1, 32);
  v += __shfl_xor(v, 2, 32);
  v += __shfl_xor(v, 4, 32);
  v += __shfl_xor(v, 8, 32);
  return v;
}
__device__ __forceinline__ float rsum32(float v) {
  v = rsum16(v);
  v += __shfl_xor(v, 16, 32);
  return v;
}

// Async global->LDS stage of one 16x128 f32 tile (8 KB) using
// GLOBAL_LOAD_ASYNC_TO_LDS_B128 (GVS mode: SGPR base + VGPR byte offset).
// Low 32 bits of a flat LDS pointer are the LDS byte offset (aperture scheme).
__device__ __forceinline__ void stage_tile_async(const float* gbase, float* As, int tid) {
  uint32_t lbase = (uint32_t)(uintptr_t)As;
  #pragma unroll
  for (int i = 0; i < 2; ++i) {
    uint32_t boff = (uint32_t)(tid * 16 + i * 4096);   // bytes
    uint32_t ldst = lbase + boff;
    asm volatile("global_load_async_to_lds_b128 %0, %1, %2"
                 :: "v"(ldst), "v"(boff), "s"(gbase) : "memory");
  }
  asm volatile("s_wait_asynccnt 0" ::: "memory");
}

// ---------------- K1: LayerNorm -> xn ----------------
__global__ void ln_kernel(const float* __restrict__ x, const float* __restrict__ gamma,
                          const float* __restrict__ beta, float* __restrict__ xn) {
  int wave = threadIdx.x >> 5, lane = threadIdx.x & 31;
  long long tok = (long long)blockIdx.x * 8 + wave;
  const float4 v = *(const float4*)(x + tok * DMODEL + lane * 4);
  float s = v.x + v.y + v.z + v.w;
  float qq = v.x*v.x + v.y*v.y + v.z*v.z + v.w*v.w;
  s = rsum32(s); qq = rsum32(qq);
  float mu = s * (1.0f / DMODEL);
  float var = qq * (1.0f / DMODEL) - mu * mu;
  float inv = rsqrtf(var + 1e-5f);
  int c = lane * 4;
  const float4 g  = *(const float4*)(gamma + c);
  const float4 bt = *(const float4*)(beta + c);
  float4 o;
  o.x = (v.x - mu) * inv * g.x + bt.x;
  o.y = (v.y - mu) * inv * g.y + bt.y;
  o.z = (v.z - mu) * inv * g.z + bt.z;
  o.w = (v.w - mu) * inv * g.w + bt.w;
  *(float4*)(xn + tok * DMODEL + c) = o;
}

// ---------------- K2: QKV GEMM (xn[65536x128] @ wqkv[128x384]) ----------------
__global__ void qkv_kernel(const float* __restrict__ xn, const float* __restrict__ wqkv,
                           float* __restrict__ q, float* __restrict__ k, float* __restrict__ v) {
  __shared__ float As[16 * 128];
  int tid = threadIdx.x;
  long long rt = blockIdx.x;
  stage_tile_async(xn + rt * 2048, As, tid);
  __syncthreads();
  int wave = tid >> 5, lane = tid & 31, l16 = lane & 15, hf = lane >> 4;
  #pragma unroll
  for (int i = 0; i < 3; ++i) {
    int tn = wave * 3 + i;
    v8f acc = {0,0,0,0,0,0,0,0};
    for (int kk = 0; kk < 32; ++kk) {
      int k0 = kk * 4 + hf * 2;
      v2f a; a.x = As[l16 * 128 + k0]; a.y = As[l16 * 128 + k0 + 1];
      v2f bf; bf.x = wqkv[k0 * 384 + tn * 16 + l16]; bf.y = wqkv[(k0 + 1) * 384 + tn * 16 + l16];
      acc = wmma4(a, bf, acc);
    }
    int col0 = tn * 16;
    int which = col0 >> 7;
    int h = (col0 & 127) >> 4;
    float scale = (which == 0) ? 0.25f : 1.0f;  // dh^-0.5
    float* dst = (which == 0) ? q : ((which == 1) ? k : v);
    #pragma unroll
    for (int r = 0; r < 8; ++r) {
      long long gt = rt * 16 + r + hf * 8;
      int b = (int)(gt >> 13), t = (int)(gt & 8191);
      dst[(((long long)(b * HEADS + h)) * NTOK + t) * DH + l16] = acc[r] * scale;
    }
  }
}

// ---------------- K3: landmark means ----------------
__global__ void landmark_kernel(const float* __restrict__ q, const float* __restrict__ k,
                                float* __restrict__ ql, float* __restrict__ kl) {
  int g = blockIdx.x * 256 + threadIdx.x;        // (bh,64,16) flat
  int bh = g >> 10, m = (g >> 4) & 63, c = g & 15;
  const float* qp = q + ((long long)bh * NTOK + m * GRP) * DH + c;
  const float* kp = k + ((long long)bh * NTOK + m * GRP) * DH + c;
  float sq = 0.f, sk = 0.f;
  for (int i = 0; i < GRP; ++i) { sq += qp[i * DH]; sk += kp[i * DH]; }
  ql[g] = sq * (1.0f / GRP);
  kl[g] = sk * (1.0f / GRP);
}

// ---------------- K4: attn2 = softmax(q_l @ k_l^T) + pinv-init stats ----------------
__global__ void sim2_kernel(const float* __restrict__ ql, const float* __restrict__ kl,
                            float* __restrict__ a2, float* __restrict__ st) {
  __shared__ float qs[1024], ks_[1024], colsum[64];
  int bh = blockIdx.x, tid = threadIdx.x;        // 128 threads
  #pragma unroll
  for (int i = 0; i < 8; ++i) {
    qs[tid + i * 128] = ql[bh * 1024 + tid + i * 128];
    ks_[tid + i * 128] = kl[bh * 1024 + tid + i * 128];
  }
  if (tid < 64) colsum[tid] = 0.f;
  __syncthreads();
  int w = tid >> 5, lane = tid & 31, l16 = lane & 15, hf = lane >> 4;
  v8f S[4];
  #pragma unroll
  for (int ct = 0; ct < 4; ++ct) {
    v8f acc = {0,0,0,0,0,0,0,0};
    #pragma unroll
    for (int kk = 0; kk < 4; ++kk) {
      int k0 = kk * 4 + hf * 2;
      v2f a; a.x = qs[(w * 16 + l16) * 16 + k0]; a.y = qs[(w * 16 + l16) * 16 + k0 + 1];
      v2f b; b.x = ks_[(ct * 16 + l16) * 16 + k0]; b.y = ks_[(ct * 16 + l16) * 16 + k0 + 1];
      acc = wmma4(a, b, acc);
    }
    S[ct] = acc;
  }
  #pragma unroll
  for (int r = 0; r < 8; ++r) {
    float mx = fmaxf(fmaxf(S[0][r], S[1][r]), fmaxf(S[2][r], S[3][r]));
    mx = rmax16(mx);
    #pragma unroll
    for (int ct = 0; ct < 4; ++ct) S[ct][r] = __expf(S[ct][r] - mx);
    float sm = S[0][r] + S[1][r] + S[2][r] + S[3][r];
    sm = rsum16(sm);
    float inv = 1.0f / sm;
    #pragma unroll
    for (int ct = 0; ct < 4; ++ct) S[ct][r] *= inv;
  }
  #pragma unroll
  for (int ct = 0; ct < 4; ++ct)
    #pragma unroll
    for (int r = 0; r < 8; ++r) {
      int row = w * 16 + r + hf * 8, col = ct * 16 + l16;
      float p = S[ct][r];
      a2[bh * 4096 + row * 64 + col] = p;
      atomicAdd(&colsum[col], p);
    }
  __syncthreads();
  if (tid == 0) {
    float m = 0.f;
    for (int i = 0; i < 64; ++i) m = fmaxf(m, colsum[i]);
    st[bh] = m;                                   // max column-sum (row-sums are exactly 1)
  }
}

// ---------------- K5: Moore-Penrose pinv (Newton-Schulz, all in LDS) ----------------
__device__ __forceinline__ void mm64(const float* S1, const float* S2, float* D,
                                     int w, int lane) {
  int l16 = lane & 15, hf = lane >> 4;
  #pragma unroll
  for (int ct = 0; ct < 4; ++ct) {
    v8f acc = {0,0,0,0,0,0,0,0};
    for (int kk = 0; kk < 16; ++kk) {
      int k0 = kk * 4 + hf * 2;
      v2f a; a.x = S1[(w * 16 + l16) * 64 + k0]; a.y = S1[(w * 16 + l16) * 64 + k0 + 1];
      v2f b; b.x = S2[k0 * 64 + ct * 16 + l16];  b.y = S2[(k0 + 1) * 64 + ct * 16 + l16];
      acc = wmma4(a, b, acc);
    }
    #pragma unroll
    for (int r = 0; r < 8; ++r)
      D[(w * 16 + r + hf * 8) * 64 + ct * 16 + l16] = acc[r];
  }
}

__device__ __forceinline__ void diag_sub(float* D, float cval, const float* S, int tid) {
  #pragma unroll
  for (int i = 0; i < 32; ++i) {
    int idx = tid + i * 128;
    int r = idx >> 6, c = idx & 63;
    D[idx] = (r == c ? cval : 0.f) - S[idx];
  }
}

__global__ void pinv_kernel(const float* __restrict__ a2, const float* __restrict__ st,
                            float* __restrict__ zi) {
  __shared__ float X[4096], Z[4096], A[4096], Bm[4096], C[4096];
  int bh = blockIdx.x, tid = threadIdx.x;        // 128 threads
  float rmax = 0.f;
  for (int i = 0; i < BH; ++i) rmax = fmaxf(rmax, st[i]);
  float zscale = 1.0f / rmax;                     // col (=1) * row
  #pragma unroll
  for (int i = 0; i < 32; ++i) { int idx = tid + i * 128; X[idx] = a2[bh * 4096 + idx]; }
  __syncthreads();
  #pragma unroll
  for (int i = 0; i < 32; ++i) {
    int idx = tid + i * 128; int r = idx >> 6, c = idx & 63;
    Z[idx] = X[c * 64 + r] * zscale;              // x^T / (col*row)
  }
  __syncthreads();
  int w = tid >> 5, lane = tid & 31;
  for (int it = 0; it < 6; ++it) {
    mm64(X, Z, A, w, lane);   __syncthreads();    // A = X@Z
    diag_sub(Bm, 7.f, A, tid);  __syncthreads();  // B = 7I - A
    mm64(A, Bm, C, w, lane);  __syncthreads();    // C = A@B
    diag_sub(Bm, 15.f, C, tid); __syncthreads();  // B = 15I - C
    mm64(A, Bm, C, w, lane);  __syncthreads();    // C = A@B
    diag_sub(Bm, 13.f, C, tid); __syncthreads();  // B = 13I - C
    mm64(Z, Bm, C, w, lane);  __syncthreads();    // C = Z@B
    #pragma unroll
    for (int i = 0; i < 32; ++i) { int idx = tid + i * 128; Z[idx] = 0.25f * C[idx]; }
    __syncthreads();
  }
  #pragma unroll
  for (int i = 0; i < 32; ++i) { int idx = tid + i * 128; zi[bh * 4096 + idx] = Z[idx]; }
}

// ---------------- K6: flash-style w2 = softmax(q_l @ k^T) @ v  (64x16 per bh) ------
__global__ void attn3v_kernel(const float* __restrict__ ql, const float* __restrict__ kg,
                              const float* __restrict__ vg, float* __restrict__ w2) {
  __shared__ float slab[8 * 1024];               // P slabs; reused for acc combine
  __shared__ float ms[8 * 64], ss[8 * 64];
  int bh = blockIdx.x, tid = threadIdx.x;
  int w = tid >> 5, lane = tid & 31, l16 = lane & 15, hf = lane >> 4;
  v2f aq[4][4];
  #pragma unroll
  for (int rt = 0; rt < 4; ++rt)
    #pragma unroll
    for (int kk = 0; kk < 4; ++kk) {
      int k0 = kk * 4 + hf * 2;
      aq[rt][kk].x = ql[bh * 1024 + (rt * 16 + l16) * 16 + k0];
      aq[rt][kk].y = ql[bh * 1024 + (rt * 16 + l16) * 16 + k0 + 1];
    }
  v8f acc[4]; float m_[4][8], s_[4][8];
  #pragma unroll
  for (int rt = 0; rt < 4; ++rt) {
    acc[rt] = (v8f){0,0,0,0,0,0,0,0};
    #pragma unroll
    for (int r = 0; r < 8; ++r) { m_[rt][r] = -1e30f; s_[rt][r] = 0.f; }
  }
  const float* kb = kg + (long long)bh * NTOK * DH;
  const float* vb = vg + (long long)bh * NTOK * DH;
  float* ps = slab + w * 1024;
  for (int ch = 0; ch < 64; ++ch) {
    int base = w * 1024 + ch * 16;               // this wave's key chunk
    v2f bk[4];
    #pragma unroll
    for (int kk = 0; kk < 4; ++kk) {
      int k0 = kk * 4 + hf * 2;
      bk[kk].x = kb[(base + l16) * DH + k0];
      bk[kk].y = kb[(base + l16) * DH + k0 + 1];
    }
    v8f S[4];
    #pragma unroll
    for (int rt = 0; rt < 4; ++rt) {
      v8f t = {0,0,0,0,0,0,0,0};
      #pragma unroll
      for (int kk = 0; kk < 4; ++kk) t = wmma4(aq[rt][kk], bk[kk], t);
      S[rt] = t;
    }
    #pragma unroll
    for (int rt = 0; rt < 4; ++rt)
      #pragma unroll
      for (int r = 0; r < 8; ++r) {
        float cm = rmax16(S[rt][r]);
        float nm = fmaxf(m_[rt][r], cm);
        float sc = __expf(m_[rt][r] - nm);
        S[rt][r] = __expf(S[rt][r] - nm);
        float rs = rsum16(S[rt][r]);
        s_[rt][r] = s_[rt][r] * sc + rs;
        acc[rt][r] *= sc;
        m_[rt][r] = nm;
      }
    #pragma unroll
    for (int rt = 0; rt < 4; ++rt)
      #pragma unroll
      for (int r = 0; r < 8; ++r)
        ps[(rt * 16 + r + hf * 8) * 16 + l16] = S[rt][r];   // wave-private, in-order DS
    v2f bv[4];
    #pragma unroll
    for (int kk = 0; kk < 4; ++kk) {
      int k0 = kk * 4 + hf * 2;
      bv[kk].x = vb[(base + k0) * DH + l16];
      bv[kk].y = vb[(base + k0 + 1) * DH + l16];
    }
    #pragma unroll
    for (int rt = 0; rt < 4; ++rt) {
      v8f t = acc[rt];
      #pragma unroll
      for (int kk = 0; kk < 4; ++kk) {
        int k0 = kk * 4 + hf * 2;
        v2f ap; ap.x = ps[(rt * 16 + l16) * 16 + k0]; ap.y = ps[(rt * 16 + l16) * 16 + k0 + 1];
        t = wmma4(ap, bv[kk], t);
      }
      acc[rt] = t;
    }
  }
  if (l16 == 0)
    #pragma unroll
    for (int rt = 0; rt < 4; ++rt)
      #pragma unroll
      for (int r = 0; r < 8; ++r) {
        int row = rt * 16 + r + hf * 8;
        ms[w * 64 + row] = m_[rt][r];
        ss[w * 64 + row] = s_[rt][r];
      }
  __syncthreads();
  float* ac = slab;                               // reuse P slabs for partial accs
  #pragma unroll
  for (int rt = 0; rt < 4; ++rt)
    #pragma unroll
    for (int r = 0; r < 8; ++r)
      ac[w * 1024 + (rt * 16 + r + hf * 8) * 16 + l16] = acc[rt][r];
  __syncthreads();
  #pragma unroll
  for (int i = 0; i < 4; ++i) {
    int idx = tid + i * 256;                      // row*16+col
    int row = idx >> 4;
    float M = -1e30f;
    for (int ww = 0; ww < 8; ++ww) M = fmaxf(M, ms[ww * 64 + row]);
    float den = 0.f, num = 0.f;
    for (int ww = 0; ww < 8; ++ww) {
      float e = __expf(ms[ww * 64 + row] - M);
      den += e * ss[ww * 64 + row];
      num += e * ac[ww * 1024 + idx];
    }
    w2[bh * 1024 + idx] = num / den;
  }
}

// ---------------- K6b: zw = attn2_inv @ w2  (64x16 per bh) ----------------
__global__ void zw_kernel(const float* __restrict__ zi, const float* __restrict__ w2,
                          float* __restrict__ zw) {
  int bh = blockIdx.x, tid = threadIdx.x;        // 128 threads
  int w = tid >> 5, lane = tid & 31, l16 = lane & 15, hf = lane >> 4;
  const float* Zp = zi + bh * 4096;
  const float* Wp = w2 + bh * 1024;
  v8f acc = {0,0,0,0,0,0,0,0};
  for (int kk = 0; kk < 16; ++kk) {
    int k0 = kk * 4 + hf * 2;
    v2f a; a.x = Zp[(w * 16 + l16) * 64 + k0]; a.y = Zp[(w * 16 + l16) * 64 + k0 + 1];
    v2f b; b.x = Wp[k0 * 16 + l16];            b.y = Wp[(k0 + 1) * 16 + l16];
    acc = wmma4(a, b, acc);
  }
  #pragma unroll
  for (int r = 0; r < 8; ++r)
    zw[bh * 1024 + (w * 16 + r + hf * 8) * 16 + l16] = acc[r];
}

// ---------------- K7: out = softmax(q @ k_l^T) @ zw + depthwise-conv(v) -> oh ------
__global__ void out_kernel(const float* __restrict__ q, const float* __restrict__ kl,
                           const float* __restrict__ zw, const float* __restrict__ vg,
                           const float* __restrict__ cw, float* __restrict__ oh) {
  __shared__ float slab[8 * 1024];               // 16x64 P per wave
  int tid = threadIdx.x, w = tid >> 5, lane = tid & 31, l16 = lane & 15, hf = lane >> 4;
  long long job = (long long)blockIdx.x * 8 + w;
  int bh = (int)(job >> 9), tile = (int)(job & 511);
  const float* qb = q + (long long)bh * NTOK * DH;
  v2f aq[4];
  #pragma unroll
  for (int kk = 0; kk < 4; ++kk) {
    int k0 = kk * 4 + hf * 2;
    aq[kk].x = qb[(tile * 16 + l16) * DH + k0];
    aq[kk].y = qb[(tile * 16 + l16) * DH + k0 + 1];
  }
  v8f S[4];
  #pragma unroll
  for (int ct = 0; ct < 4; ++ct) {
    v8f t = {0,0,0,0,0,0,0,0};
    #pragma unroll
    for (int kk = 0; kk < 4; ++kk) {
      int k0 = kk * 4 + hf * 2;
      v2f b; b.x = kl[bh * 1024 + (ct * 16 + l16) * 16 + k0];
             b.y = kl[bh * 1024 + (ct * 16 + l16) * 16 + k0 + 1];
      t = wmma4(aq[kk], b, t);
    }
    S[ct] = t;
  }
  #pragma unroll
  for (int r = 0; r < 8; ++r) {
    float mx = fmaxf(fmaxf(S[0][r], S[1][r]), fmaxf(S[2][r], S[3][r]));
    mx = rmax16(mx);
    #pragma unroll
    for (int ct = 0; ct < 4; ++ct) S[ct][r] = __expf(S[ct][r] - mx);
    float sm = S[0][r] + S[1][r] + S[2][r] + S[3][r];
    sm = rsum16(sm);
    float inv = 1.0f / sm;
    #pragma unroll
    for (int ct = 0; ct < 4; ++ct) S[ct][r] *= inv;
  }
  float* ps = slab + w * 1024;
  #pragma unroll
  for (int ct = 0; ct < 4; ++ct)
    #pragma unroll
    for (int r = 0; r < 8; ++r)
      ps[(r + hf * 8) * 64 + ct * 16 + l16] = S[ct][r];
  v8f o = {0,0,0,0,0,0,0,0};
  for (int kk = 0; kk < 16; ++kk) {
    int k0 = kk * 4 + hf * 2;
    v2f a; a.x = ps[l16 * 64 + k0]; a.y = ps[l16 * 64 + k0 + 1];
    v2f b; b.x = zw[bh * 1024 + k0 * 16 + l16]; b.y = zw[bh * 1024 + (k0 + 1) * 16 + l16];
    o = wmma4(a, b, o);
  }
  int b_ = bh >> 3, h_ = bh & 7;
  const float* vb = vg + (long long)bh * NTOK * DH;
  float wts[KSZ];
  #pragma unroll
  for (int t = 0; t < KSZ; ++t) wts[t] = cw[h_ * KSZ + t];
  #pragma unroll
  for (int r = 0; r < 8; ++r) {
    int t = tile * 16 + r + hf * 8;
    float res = 0.f;
    for (int tap = 0; tap < KSZ; ++tap) {
      int tt = t + tap - 16;
      if (tt >= 0 && tt < NTOK) res += wts[tap] * vb[tt * DH + l16];
    }
    oh[((long long)b_ * NTOK + t) * DMODEL + h_ * DH + l16] = o[r] + res;
  }
}

// ---------------- K8: final projection + bias + residual ----------------
__global__ void proj_kernel(const float* __restrict__ oh, const float* __restrict__ wout,
                            const float* __restrict__ bout, const float* __restrict__ x,
                            float* __restrict__ out) {
  __shared__ float As[16 * 128];
  long long rt = blockIdx.x;
  int tid = threadIdx.x;
  stage_tile_async(oh + rt * 2048, As, tid);
  __syncthreads();
  int w = tid >> 5, lane = tid & 31, l16 = lane & 15, hf = lane >> 4;
  int col = w * 16 + l16;
  v8f acc = {0,0,0,0,0,0,0,0};
  for (int kk = 0; kk < 32; ++kk) {
    int k0 = kk * 4 + hf * 2;
    v2f a; a.x = As[l16 * 128 + k0]; a.y = As[l16 * 128 + k0 + 1];
    v2f b; b.x = wout[k0 * 128 + col]; b.y = wout[(k0 + 1) * 128 + col];
    acc = wmma4(a, b, acc);
  }
  float bb = bout[col];
  #pragma unroll
  for (int r = 0; r < 8; ++r) {
    long long gt = rt * 16 + r + hf * 8;
    out[gt * 128 + col] = acc[r] + bb + x[gt * 128 + col];
  }
}

extern "C" void kernel_launch(void* const* d_in, const int* in_sizes, int n_in,
                              void* d_out, int out_size, void* d_ws, size_t ws_size,
                              hipStream_t stream) {
  const float* x     = (const float*)d_in[0];
  const float* gamma = (const float*)d_in[1];
  const float* beta  = (const float*)d_in[2];
  const float* wqkv  = (const float*)d_in[3];
  const float* wout  = (const float*)d_in[4];
  const float* bout  = (const float*)d_in[5];
  const float* cw    = (const float*)d_in[6];
  float* ws = (float*)d_ws;
  size_t o = 0;
  float* xn = ws + o; o += (size_t)ROWS * DMODEL;       // reused as oh later
  float* q  = ws + o; o += (size_t)BH * NTOK * DH;
  float* k  = ws + o; o += (size_t)BH * NTOK * DH;
  float* v  = ws + o; o += (size_t)BH * NTOK * DH;
  float* ql = ws + o; o += (size_t)BH * LM * DH;
  float* kl = ws + o; o += (size_t)BH * LM * DH;
  float* a2 = ws + o; o += (size_t)BH * LM * LM;
  float* zi = ws + o; o += (size_t)BH * LM * LM;
  float* w2 = ws + o; o += (size_t)BH * LM * DH;
  float* zw = ws + o; o += (size_t)BH * LM * DH;
  float* st = ws + o; o += (size_t)BH;
  float* oh = xn;

  ln_kernel      <<<ROWS / 8, 256, 0, stream>>>(x, gamma, beta, xn);
  qkv_kernel     <<<ROWS / 16, 256, 0, stream>>>(xn, wqkv, q, k, v);
  landmark_kernel<<<(BH * LM * DH) / 256, 256, 0, stream>>>(q, k, ql, kl);
  sim2_kernel    <<<BH, 128, 0, stream>>>(ql, kl, a2, st);
  pinv_kernel    <<<BH, 128, 0, stream>>>(a2, st, zi);
  attn3v_kernel  <<<BH, 256, 0, stream>>>(ql, k, v, w2);
  zw_kernel      <<<BH, 128, 0, stream>>>(zi, w2, zw);
  out_kernel     <<<(BH * 512) / 8, 256, 0, stream>>>(q, kl, zw, v, cw, oh);
  proj_kernel    <<<ROWS / 16, 256, 0, stream>>>(oh, wout, bout, x, (float*)d_out);
}